// OneLayerGCN_5566277615674
// MI455X (gfx1250) — compile-verified
//
#include <hip/hip_runtime.h>
#include <hip/hip_bf16.h>

typedef float v2f __attribute__((ext_vector_type(2)));
typedef float v8f __attribute__((ext_vector_type(8)));

#define D_FEAT 64  // D_IN == D_OUT == 64 in the reference

// ---------------------------------------------------------------------------
// Phase 1: degree with self loops.  deg starts at 1.0 (the self loop), then
// one atomic add per edge on the target node, then dinv = rsqrt(deg) in place.
// ---------------------------------------------------------------------------
__global__ void gcn_init_deg(float* __restrict__ deg, int n) {
  int i = blockIdx.x * blockDim.x + threadIdx.x;
  if (i < n) deg[i] = 1.0f;
}

__global__ void gcn_deg_accum(const long long* __restrict__ col,
                              float* __restrict__ deg, int nedges) {
  int i = blockIdx.x * blockDim.x + threadIdx.x;
  if (i < nedges) atomicAdd(&deg[(int)col[i]], 1.0f);
}

__global__ void gcn_deg_to_dinv(float* __restrict__ deg, int n) {
  int i = blockIdx.x * blockDim.x + threadIdx.x;
  if (i < n) deg[i] = __frsqrt_rn(deg[i]);  // deg >= 1 always (self loop)
}

// ---------------------------------------------------------------------------
// Phase 2: xw = x @ W via V_WMMA_F32_16X16X4_F32 (fp32, matches reference).
// 256-thread block = 8 waves; wave w owns the 16x16 tile at
//   rows: (2*blockIdx.x + w/4)*16 , cols: (w%4)*16.
// K=64 -> 16 WMMA issues per tile.
// Epilogue fuses the self-loop term: out = c * dinv[row]^2 + b, and also
// writes xw to workspace for the edge-scatter phase.
//
// Fragment layouts per CDNA5 ISA 7.12.2 (wave32):
//   A (16x4 f32, 2 VGPR):  lanes 0-15: M=lane, (K=k, k+1); lanes 16-31: M=lane-16, (K=k+2, k+3)
//   B (4x16  f32, 2 VGPR): lanes 0-15: N=lane, rows (K=k, k+1); lanes 16-31: rows (K=k+2, k+3)
//   C/D (16x16 f32, 8 VGPR): vgpr i: lanes 0-15 -> M=i, N=lane; lanes 16-31 -> M=i+8
// ---------------------------------------------------------------------------
__global__ __launch_bounds__(256) void gcn_gemm_wmma(
    const float* __restrict__ x, const float* __restrict__ W,
    const float* __restrict__ b, const float* __restrict__ dinv,
    float* __restrict__ xw, float* __restrict__ out, int n_nodes) {
  const int lane = threadIdx.x & 31;
  const int wave = threadIdx.x >> 5;            // 0..7
  const int hi   = lane >> 4;                   // half-wave: 0 or 1
  const int l    = lane & 15;

  const int row0 = (blockIdx.x * 2 + (wave >> 2)) * 16;  // 16-row tile base
  const int n0   = (wave & 3) * 16;                      // 16-col tile base

  // Clamp A-row so all 32 lanes stay active (WMMA requires EXEC all-ones).
  int arow = row0 + l;
  if (arow >= n_nodes) arow = n_nodes - 1;

  v8f c = {};
#pragma unroll
  for (int k = 0; k < D_FEAT; k += 4) {
    // A fragment: this lane's row, K = k + 2*hi + {0,1}  (8-byte load)
    const float* ap = x + (size_t)arow * D_FEAT + k + 2 * hi;
    v2f a;
    a.x = ap[0];
    a.y = ap[1];
    // B fragment: rows K = k + 2*hi + {0,1}, column n0 + l
    const float* bp = W + (size_t)(k + 2 * hi) * D_FEAT + n0 + l;
    v2f bb;
    bb.x = bp[0];
    bb.y = bp[D_FEAT];
    // 8 args: (neg_a, A, neg_b, B, c_mod, C, reuse_a, reuse_b)
    c = __builtin_amdgcn_wmma_f32_16x16x4_f32(false, a, false, bb, (short)0, c,
                                              false, false);
  }

  const int col = n0 + l;
  const float bv = b[col];
#pragma unroll
  for (int i = 0; i < 8; ++i) {
    const int r = row0 + i + hi * 8;
    if (r < n_nodes) {
      const float v  = c[i];
      const float di = dinv[r];
      xw[(size_t)r * D_FEAT + col]  = v;              // for edge scatter
      out[(size_t)r * D_FEAT + col] = v * di * di + bv;  // self-loop + bias
    }
  }
}

// ---------------------------------------------------------------------------
// Phase 3: per-edge scatter.  One 32-lane wave per edge; each lane handles 2
// features (float2 gather from xw -> 2 f32 atomics into out).  All working
// data (xw, out, dinv) fits in the 192 MB L2, so gathers/atomics stay on-chip.
// ---------------------------------------------------------------------------
__global__ __launch_bounds__(256) void gcn_edge_scatter(
    const long long* __restrict__ rowi, const long long* __restrict__ coli,
    const float* __restrict__ dinv, const float* __restrict__ xw,
    float* __restrict__ out, int nedges) {
  const long long e = (long long)blockIdx.x * 8 + (threadIdx.x >> 5);
  if (e >= nedges) return;
  const int f = (threadIdx.x & 31) * 2;

  const int r  = (int)rowi[e];
  const int ct = (int)coli[e];
  const float nm = dinv[r] * dinv[ct];

  const float2 v = *(const float2*)(xw + (size_t)r * D_FEAT + f);
  float* dst = out + (size_t)ct * D_FEAT + f;
  atomicAdd(dst + 0, v.x * nm);
  atomicAdd(dst + 1, v.y * nm);
}

// ---------------------------------------------------------------------------
// Launcher.  Inputs (setup_inputs order): x [N*64 f32], edge_index [2*E i64],
// W [64*64 f32], b [64 f32].  Output: [N*64 f32].
// Workspace: dinv (N f32) then xw (N*64 f32)  ~= 26 MB.
// ---------------------------------------------------------------------------
extern "C" void kernel_launch(void* const* d_in, const int* in_sizes, int n_in,
                              void* d_out, int out_size, void* d_ws,
                              size_t ws_size, hipStream_t stream) {
  const float* x            = (const float*)d_in[0];
  const long long* eidx     = (const long long*)d_in[1];  // int64 per reference
  const float* W            = (const float*)d_in[2];
  const float* b            = (const float*)d_in[3];
  float* out                = (float*)d_out;

  const int n_nodes = in_sizes[0] / D_FEAT;
  const int n_edges = in_sizes[1] / 2;
  const long long* rowi = eidx;            // edge_index[0]: sources
  const long long* coli = eidx + n_edges;  // edge_index[1]: targets

  float* dinv = (float*)d_ws;
  size_t dinv_elems = ((size_t)n_nodes + 255) & ~(size_t)255;
  float* xw = dinv + dinv_elems;

  // Phase 1: degrees -> dinv
  {
    int tb = 256;
    gcn_init_deg<<<(n_nodes + tb - 1) / tb, tb, 0, stream>>>(dinv, n_nodes);
    gcn_deg_accum<<<(n_edges + tb - 1) / tb, tb, 0, stream>>>(coli, dinv,
                                                              n_edges);
    gcn_deg_to_dinv<<<(n_nodes + tb - 1) / tb, tb, 0, stream>>>(dinv, n_nodes);
  }

  // Phase 2: WMMA GEMM + fused self-loop/bias epilogue (32 rows per block)
  {
    int blocks = (n_nodes + 31) / 32;
    gcn_gemm_wmma<<<blocks, 256, 0, stream>>>(x, W, b, dinv, xw, out, n_nodes);
  }

  // Phase 3: edge scatter (8 edges per 256-thread block)
  {
    int blocks = (n_edges + 7) / 8;
    gcn_edge_scatter<<<blocks, 256, 0, stream>>>(rowi, coli, dinv, xw, out,
                                                 n_edges);
  }
}